// GraphSAGE_27350351741495
// MI455X (gfx1250) — compile-verified
//
#include <hip/hip_runtime.h>
#include <math.h>

typedef __attribute__((ext_vector_type(2))) float v2f;
typedef __attribute__((ext_vector_type(4))) float v4f;
typedef __attribute__((ext_vector_type(8))) float v8f;

// ---------------------------------------------------------------------------
// Row-wise argmax over a [n, n] fp32 matrix. One block per row, coalesced
// non-temporal b128 streaming (this is the bandwidth-dominant 800MB phase;
// NT hint keeps the one-shot stream from cycling L2, which we want to keep
// for the scatter accumulators). Tie-break: smallest index (jnp.argmax).
// ---------------------------------------------------------------------------
__global__ __launch_bounds__(256) void row_argmax_kernel(const float* __restrict__ logits,
                                                         int* __restrict__ perm, int n) {
  __shared__ float sval[256];
  __shared__ int   sidx[256];
  const int row = blockIdx.x;
  const float* r = logits + (size_t)row * n;
  float best = -__builtin_inff();
  int bidx = 0;
  const int nvec = n >> 2;
  const v4f* r4 = (const v4f*)r;
  for (int i = threadIdx.x; i < nvec; i += 256) {
    v4f v = __builtin_nontemporal_load(r4 + i);   // streaming NT load
    const int base = i << 2;
    if (v.x > best) { best = v.x; bidx = base; }
    if (v.y > best) { best = v.y; bidx = base + 1; }
    if (v.z > best) { best = v.z; bidx = base + 2; }
    if (v.w > best) { best = v.w; bidx = base + 3; }
  }
  for (int i = (nvec << 2) + threadIdx.x; i < n; i += 256) {
    float v = r[i];
    if (v > best) { best = v; bidx = i; }
  }
  sval[threadIdx.x] = best; sidx[threadIdx.x] = bidx;
  __syncthreads();
  for (int s = 128; s > 0; s >>= 1) {
    if (threadIdx.x < s) {
      float ov = sval[threadIdx.x + s]; int oi = sidx[threadIdx.x + s];
      float mv = sval[threadIdx.x];     int mi = sidx[threadIdx.x];
      if (ov > mv || (ov == mv && oi < mi)) { sval[threadIdx.x] = ov; sidx[threadIdx.x] = oi; }
    }
    __syncthreads();
  }
  if (threadIdx.x == 0) perm[row] = sidx[0];
}

// ---------------------------------------------------------------------------
// Edge scatter-mean numerator + counts. One wave32 per edge; each lane owns
// F/32 contiguous features. Accumulator (~5MB) lives in L2; fp32 atomics.
// ---------------------------------------------------------------------------
template<int F>
__global__ __launch_bounds__(256) void scatter_mean_kernel(const float* __restrict__ x,
    const int* __restrict__ ei, const int* __restrict__ perm,
    float* __restrict__ agg, float* __restrict__ cnt, int nE) {
  const int wid  = blockIdx.x * (blockDim.x >> 5) + (threadIdx.x >> 5);
  const int lane = threadIdx.x & 31;
  if (wid >= nE) return;
  const int s = perm[ei[wid]];        // src = perm[edge_index[0][e]]
  const int t = perm[ei[nE + wid]];   // tgt = perm[edge_index[1][e]]
  constexpr int PER = F / 32;
  const float* xr = x + (size_t)s * F + lane * PER;
  float* ar = agg + (size_t)t * F + lane * PER;
  #pragma unroll
  for (int j = 0; j < PER; ++j) atomicAdd(ar + j, xr[j]);
  if (lane == 0) atomicAdd(cnt + t, 1.0f);
}

template<int F>
__global__ __launch_bounds__(256) void normalize_kernel(float* __restrict__ agg,
                                                        const float* __restrict__ cnt, int n) {
  const int i = blockIdx.x * blockDim.x + threadIdx.x;
  if (i >= n * F) return;
  const float c = cnt[i / F];
  agg[i] *= 1.0f / fmaxf(c, 1.0f);
}

// ---------------------------------------------------------------------------
// Zero-pad layer-2 weights [40,K] -> [48,K] and bias [40] -> [48] once per
// call, so the WMMA GEMM needs no per-element masking at all.
// ---------------------------------------------------------------------------
__global__ __launch_bounds__(256) void pad_weights_kernel(
    const float* __restrict__ W1, const float* __restrict__ W2,
    const float* __restrict__ b,
    float* __restrict__ W1p, float* __restrict__ W2p, float* __restrict__ bp,
    int O, int K, int OPad) {
  const int i = blockIdx.x * blockDim.x + threadIdx.x;
  const int total = OPad * K;
  if (i < total) {
    const int o = i / K, k = i - o * K;
    const int co = (o < O) ? o : 0;
    const float m = (o < O) ? 1.0f : 0.0f;
    W1p[i] = W1[co * K + k] * m;
    W2p[i] = W2[co * K + k] * m;
  }
  if (i < OPad) bp[i] = ((i < O) ? b[i] : 0.0f);
}

// ---------------------------------------------------------------------------
// Fused SAGE linear: out = relu(agg @ W1^T + bias + x @ W2^T) via
// V_WMMA_F32_16X16X4_F32. One wave per 16x16 output tile; both GEMMs share the
// 8-VGPR f32 accumulator. A layout: lanes0-15 M=0..15 K={0,1}, lanes16-31
// K={2,3}; B mirrors with N across lanes. C/D: VGPR r = rows r / r+8.
// Weights are pre-padded, so the inner loop is purely unconditional
// 8B-aligned b64 loads + WMMA; EXEC stays all-ones as the ISA requires.
// ---------------------------------------------------------------------------
template<int K, int OUT_TILES>
__global__ __launch_bounds__(256) void sage_linear_wmma(
    const float* __restrict__ A1,   // agg  [nRows, K]
    const float* __restrict__ A2,   // x/h  [nRows, K]
    const float* __restrict__ W1,   // [OUT_TILES*16, K] row-major (padded)
    const float* __restrict__ W2,   // [OUT_TILES*16, K] row-major (padded)
    const float* __restrict__ bias, // [OUT_TILES*16] (padded)
    float* __restrict__ out,        // [nRows, OPad]
    int nRows, int OPad) {
  const int wid  = blockIdx.x * (blockDim.x >> 5) + (threadIdx.x >> 5);
  const int lane = threadIdx.x & 31;
  const int mTiles = nRows >> 4;
  if (wid >= mTiles * OUT_TILES) return;          // wave-uniform exit
  const int tm = (wid / OUT_TILES) << 4;
  const int tn = (wid % OUT_TILES) << 4;

  const int half = lane >> 4;                     // 0: lanes0-15, 1: lanes16-31
  const int lid  = lane & 15;
  const int koff = half << 1;                     // K sub-offset 0 or 2

  const float* a1row = A1 + (size_t)(tm + lid) * K + koff;
  const float* a2row = A2 + (size_t)(tm + lid) * K + koff;
  const int ocol = tn + lid;
  const float* w1row = W1 + (size_t)ocol * K + koff;
  const float* w2row = W2 + (size_t)ocol * K + koff;

  v8f c = {};
  #pragma unroll
  for (int k = 0; k < K; k += 4) {
    v2f a1 = *(const v2f*)(a1row + k);
    v2f b1 = *(const v2f*)(w1row + k);
    c = __builtin_amdgcn_wmma_f32_16x16x4_f32(false, a1, false, b1, (short)0, c, false, false);
    v2f a2 = *(const v2f*)(a2row + k);
    v2f b2 = *(const v2f*)(w2row + k);
    c = __builtin_amdgcn_wmma_f32_16x16x4_f32(false, a2, false, b2, (short)0, c, false, false);
  }

  const float bv = bias[ocol];
  #pragma unroll
  for (int r = 0; r < 8; ++r) {
    const int row = tm + r + (half << 3);
    const float v = c[r] + bv;
    out[(size_t)row * OPad + ocol] = fmaxf(v, 0.0f);   // ReLU (both layers)
  }
}

// ---------------------------------------------------------------------------
// log_softmax over C=40 classes (row stride padded to 48). One thread per row.
// ---------------------------------------------------------------------------
__global__ __launch_bounds__(256) void log_softmax_kernel(const float* __restrict__ h,
                                                          float* __restrict__ out,
                                                          int n, int C, int pad) {
  const int i = blockIdx.x * blockDim.x + threadIdx.x;
  if (i >= n) return;
  const float* r = h + (size_t)i * pad;
  float m = -__builtin_inff();
  for (int j = 0; j < C; ++j) m = fmaxf(m, r[j]);
  float s = 0.0f;
  for (int j = 0; j < C; ++j) s += __expf(r[j] - m);
  const float ls = __logf(s);
  float* o = out + (size_t)i * C;
  for (int j = 0; j < C; ++j) o[j] = r[j] - m - ls;
}

extern "C" void kernel_launch(void* const* d_in, const int* in_sizes, int n_in,
                              void* d_out, int out_size, void* d_ws, size_t ws_size,
                              hipStream_t stream) {
  (void)n_in; (void)out_size; (void)ws_size;
  const float* x       = (const float*)d_in[0];
  const int*   ei      = (const int*)d_in[1];   // [2, E] flat
  const float* logits0 = (const float*)d_in[2];
  const float* Wl0     = (const float*)d_in[3];
  const float* Wr0     = (const float*)d_in[4];
  const float* b0      = (const float*)d_in[5];
  const float* logits1 = (const float*)d_in[6];
  const float* Wl1     = (const float*)d_in[7];
  const float* Wr1     = (const float*)d_in[8];
  const float* b1      = (const float*)d_in[9];
  float* out = (float*)d_out;

  const int N = 10000;
  const int E = in_sizes[1] / 2;

  // Workspace carve-up (256B aligned slices).
  char* ws = (char*)d_ws;
  size_t off = 0;
  auto carve = [&](size_t bytes) -> void* {
    void* p = ws + off;
    off += (bytes + 255) & ~(size_t)255;
    return p;
  };
  int*   perm0 = (int*)  carve((size_t)N * 4);
  int*   perm1 = (int*)  carve((size_t)N * 4);
  float* cnt0  = (float*)carve((size_t)N * 4);
  float* cnt1  = (float*)carve((size_t)N * 4);
  float* agg0  = (float*)carve((size_t)N * 128 * 4);
  float* h1    = (float*)carve((size_t)N * 64 * 4);
  float* agg1  = (float*)carve((size_t)N * 64 * 4);
  float* h2    = (float*)carve((size_t)N * 48 * 4);   // layer2 out, padded 40->48
  float* Wl1p  = (float*)carve((size_t)48 * 64 * 4);  // padded layer2 weights
  float* Wr1p  = (float*)carve((size_t)48 * 64 * 4);
  float* b1p   = (float*)carve((size_t)48 * 4);

  // Atomic accumulators must start at zero every call.
  hipMemsetAsync(agg0, 0, (size_t)N * 128 * 4, stream);
  hipMemsetAsync(cnt0, 0, (size_t)N * 4, stream);
  hipMemsetAsync(agg1, 0, (size_t)N * 64 * 4, stream);
  hipMemsetAsync(cnt1, 0, (size_t)N * 4, stream);

  // Zero-pad layer-2 weights/bias [40,64] -> [48,64] (tiny, overlaps argmax).
  pad_weights_kernel<<<(48 * 64 + 255) / 256, 256, 0, stream>>>(
      Wl1, Wr1, b1, Wl1p, Wr1p, b1p, 40, 64, 48);

  // perm = argmax(logits, axis=1)  (the 800MB streaming phase)
  row_argmax_kernel<<<N, 256, 0, stream>>>(logits0, perm0, N);
  row_argmax_kernel<<<N, 256, 0, stream>>>(logits1, perm1, N);

  // ---- layer 1 ----
  {
    const int blocks = (E + 7) / 8;                 // 8 waves/block, 1 wave/edge
    scatter_mean_kernel<128><<<blocks, 256, 0, stream>>>(x, ei, perm0, agg0, cnt0, E);
    normalize_kernel<128><<<(N * 128 + 255) / 256, 256, 0, stream>>>(agg0, cnt0, N);
    const int tiles = (N / 16) * 4;                 // 625 x 4 = 2500 tiles
    sage_linear_wmma<128, 4><<<(tiles + 7) / 8, 256, 0, stream>>>(
        agg0, x, Wl0, Wr0, b0, h1, N, 64);
  }

  // ---- layer 2 ----
  {
    const int blocks = (E + 7) / 8;
    scatter_mean_kernel<64><<<blocks, 256, 0, stream>>>(h1, ei, perm1, agg1, cnt1, E);
    normalize_kernel<64><<<(N * 64 + 255) / 256, 256, 0, stream>>>(agg1, cnt1, N);
    const int tiles = (N / 16) * 3;                 // 625 x 3 = 1875 tiles (48 padded cols)
    sage_linear_wmma<64, 3><<<(tiles + 7) / 8, 256, 0, stream>>>(
        agg1, h1, Wl1p, Wr1p, b1p, h2, N, 48);
  }

  // ---- log_softmax over 40 classes ----
  log_softmax_kernel<<<(N + 255) / 256, 256, 0, stream>>>(h2, out, N, 40, 48);
}